// SSIM_42872363549017
// MI455X (gfx1250) — compile-verified
//
#include <hip/hip_runtime.h>

// ---------------------------------------------------------------------------
// MS-SSIM for (32,3,512,512) f32 pairs on MI455X (gfx1250, wave32).
// Per 16x16 tile: LDS-staged 20x20 halo, VALU horizontal Gaussian pass,
// vertical Gaussian pass as banded matmul on V_WMMA_F32_16X16X4_F32,
// fused pointwise SSIM/CS + wave reduction + f64 atomic accumulation.
// ---------------------------------------------------------------------------

typedef float v2f __attribute__((ext_vector_type(2)));
typedef float v8f __attribute__((ext_vector_type(8)));

// Normalized 5-tap Gaussian, sigma = 1.5 (matches reference make_window)
__constant__ float G5[5] = {0.12007838f, 0.23388076f, 0.29208172f,
                            0.23388076f, 0.12007838f};

#define SSIM_C1 0.0001f   // 0.01^2
#define SSIM_C2 0.0009f   // 0.03^2
#define PLANES  96        // 32 batch * 3 channels

__global__ void init_acc_kernel(double* acc) {
    if (threadIdx.x < 16) acc[threadIdx.x] = 0.0;
}

__global__ __launch_bounds__(32)
void ssim_level_kernel(const float* __restrict__ img1,
                       const float* __restrict__ img2,
                       int S, double* accS, double* accC)
{
    // halo tiles (20x20) for both images; horizontal conv results H[q][20][16]
    __shared__ float sIn1[20][20];
    __shared__ float sIn2[20][20];
    __shared__ float sH[5][20][16];

    const int  lane  = threadIdx.x;            // 0..31 (wave32)
    const int  x0    = blockIdx.x * 16;
    const int  y0    = blockIdx.y * 16;
    const long plane = blockIdx.z;
    const float* p1 = img1 + plane * (long)S * (long)S;
    const float* p2 = img2 + plane * (long)S * (long)S;

    // ---- stage 20x20 halo, zero padded (matches JAX zero 'same' padding) ---
    for (int i = lane; i < 400; i += 32) {
        int r = i / 20, c = i - 20 * (i / 20);
        int gy = y0 + r - 2, gx = x0 + c - 2;
        float a = 0.0f, b = 0.0f;
        if (gy >= 0 && gy < S && gx >= 0 && gx < S) {
            long off = (long)gy * S + gx;
            a = p1[off];
            b = p2[off];
        }
        sIn1[r][c] = a;
        sIn2[r][c] = b;
    }
    __syncthreads();

    // ---- horizontal 5-tap pass on 20 rows x 16 cols, 5 quantities ---------
    for (int i = lane; i < 320; i += 32) {
        int r = i >> 4, c = i & 15;
        float h0 = 0.f, h1 = 0.f, h2 = 0.f, h3 = 0.f, h4 = 0.f;
#pragma unroll
        for (int k = 0; k < 5; ++k) {
            float g = G5[k];
            float a = sIn1[r][c + k];
            float b = sIn2[r][c + k];
            h0 += g * a;           // mu1
            h1 += g * b;           // mu2
            h2 += g * a * a;       // E[x1^2]
            h3 += g * b * b;       // E[x2^2]
            h4 += g * a * b;       // E[x1*x2]
        }
        sH[0][r][c] = h0; sH[1][r][c] = h1; sH[2][r][c] = h2;
        sH[3][r][c] = h3; sH[4][r][c] = h4;
    }
    __syncthreads();

    // ---- vertical pass: out(16x16) = Gv(16x20) x H(20x16) -----------------
    // Gv[m][j] = g[j - m] for j-m in [0,4].  K=20 -> five K=4 WMMA chunks.
    // A layout (16x4 f32): lane m: v0=K0, v1=K1; lane m+16: v0=K2, v1=K3.
    // B layout (4x16 f32): v0: lanes0-15 row K0, lanes16-31 row K2;
    //                      v1: lanes0-15 row K1, lanes16-31 row K3.
    // C/D layout (16x16 f32): lane l, vgpr v -> M = v + 8*(l/16), N = l%16.
    const int m    = lane & 15;
    const int koff = (lane >> 4) ? 2 : 0;
    const int n    = m;

    v2f aCh[5];
#pragma unroll
    for (int c2 = 0; c2 < 5; ++c2) {
        int t0 = 4 * c2 + koff - m;
        int t1 = t0 + 1;
        aCh[c2].x = (t0 >= 0 && t0 <= 4) ? G5[t0] : 0.0f;
        aCh[c2].y = (t1 >= 0 && t1 <= 4) ? G5[t1] : 0.0f;
    }

    v8f O[5];
#pragma unroll
    for (int q = 0; q < 5; ++q) {
        v8f acc = {0.f, 0.f, 0.f, 0.f, 0.f, 0.f, 0.f, 0.f};
#pragma unroll
        for (int c2 = 0; c2 < 5; ++c2) {
            int r0 = 4 * c2 + koff;
            v2f b;
            b.x = sH[q][r0][n];
            b.y = sH[q][r0 + 1][n];
            acc = __builtin_amdgcn_wmma_f32_16x16x4_f32(
                false, aCh[c2], false, b, (short)0, acc, false, false);
        }
        O[q] = acc;
    }

    // ---- pointwise SSIM / CS + per-lane sums ------------------------------
    float ssum = 0.f, csum = 0.f;
#pragma unroll
    for (int v = 0; v < 8; ++v) {
        float mu1 = O[0][v], mu2 = O[1][v];
        float e11 = O[2][v], e22 = O[3][v], e12 = O[4][v];
        float mu1sq = mu1 * mu1, mu2sq = mu2 * mu2, mu12 = mu1 * mu2;
        float s11 = e11 - mu1sq;
        float s22 = e22 - mu2sq;
        float s12 = e12 - mu12;
        float den2 = s11 + s22 + SSIM_C2;
        float num2 = 2.0f * s12 + SSIM_C2;
        float cs   = num2 / den2;
        float ssim = ((2.0f * mu12 + SSIM_C1) * num2) /
                     ((mu1sq + mu2sq + SSIM_C1) * den2);
        ssum += ssim;
        csum += cs;
    }

    // ---- wave32 butterfly reduction + f64 atomics -------------------------
    for (int off = 16; off > 0; off >>= 1) {
        ssum += __shfl_xor(ssum, off, 32);
        csum += __shfl_xor(csum, off, 32);
    }
    if (lane == 0) {
        atomicAdd(accS, (double)ssum);
        atomicAdd(accC, (double)csum);
    }
}

__global__ void downsample_kernel(const float* __restrict__ in1,
                                  const float* __restrict__ in2,
                                  float* __restrict__ out1,
                                  float* __restrict__ out2,
                                  int So, int total)
{
    int idx = blockIdx.x * blockDim.x + threadIdx.x;
    if (idx >= total) return;
    int Si = So * 2;
    int pp = So * So;
    int plane = idx / pp;
    int rem = idx - plane * pp;
    int y = rem / So;
    int x = rem - y * So;
    long ib = (long)plane * Si * Si + (long)(2 * y) * Si + 2 * x;
    out1[idx] = 0.25f * (in1[ib] + in1[ib + 1] + in1[ib + Si] + in1[ib + Si + 1]);
    out2[idx] = 0.25f * (in2[ib] + in2[ib + 1] + in2[ib + Si] + in2[ib + Si + 1]);
}

__global__ void finalize_kernel(const double* __restrict__ acc,
                                float* __restrict__ out)
{
    const double W[5]   = {0.0448, 0.2856, 0.3001, 0.2363, 0.1333};
    const double cnt[5] = {(double)PLANES * 512.0 * 512.0,
                           (double)PLANES * 256.0 * 256.0,
                           (double)PLANES * 128.0 * 128.0,
                           (double)PLANES * 64.0  * 64.0,
                           (double)PLANES * 32.0  * 32.0};
    double s[5], mcs[5];
    for (int l = 0; l < 5; ++l) {
        s[l]   = (acc[l]     / cnt[l] + 1.0) * 0.5;
        mcs[l] = (acc[8 + l] / cnt[l] + 1.0) * 0.5;
    }
    // reference: prod(pow1[:-1] * pow2[-1])  ==  prod_i (mcs_i^w_i * s4^w4)
    double p2last = pow(s[4], W[4]);
    double res = 1.0;
    for (int i = 0; i < 4; ++i) res *= pow(mcs[i], W[i]) * p2last;
    out[0] = (float)res;
}

extern "C" void kernel_launch(void* const* d_in, const int* in_sizes, int n_in,
                              void* d_out, int out_size, void* d_ws, size_t ws_size,
                              hipStream_t stream)
{
    (void)in_sizes; (void)n_in; (void)out_size; (void)ws_size;

    const float* img1 = (const float*)d_in[0];
    const float* img2 = (const float*)d_in[1];
    float* out = (float*)d_out;

    // workspace layout: 16 doubles (acc: [0..4]=ssim sums, [8..12]=cs sums)
    // then the downsampled pyramid (both images, levels 1..4) ~64MB
    double* acc = (double*)d_ws;
    float* p = (float*)d_ws + 32;  // 16 doubles == 32 floats
    const long P = PLANES;
    float* l1a = p;
    float* l1b = l1a + P * 256 * 256;
    float* l2a = l1b + P * 256 * 256;
    float* l2b = l2a + P * 128 * 128;
    float* l3a = l2b + P * 128 * 128;
    float* l3b = l3a + P * 64 * 64;
    float* l4a = l3b + P * 64 * 64;
    float* l4b = l4a + P * 32 * 32;

    init_acc_kernel<<<1, 32, 0, stream>>>(acc);

    auto ssim = [&](const float* a, const float* b, int S, int lvl) {
        dim3 grid(S / 16, S / 16, PLANES);
        ssim_level_kernel<<<grid, 32, 0, stream>>>(a, b, S, acc + lvl, acc + 8 + lvl);
    };
    auto down = [&](const float* a, const float* b, float* oa, float* ob, int So) {
        int total = PLANES * So * So;
        downsample_kernel<<<(total + 255) / 256, 256, 0, stream>>>(a, b, oa, ob, So, total);
    };

    ssim(img1, img2, 512, 0);
    down(img1, img2, l1a, l1b, 256);
    ssim(l1a, l1b, 256, 1);
    down(l1a, l1b, l2a, l2b, 128);
    ssim(l2a, l2b, 128, 2);
    down(l2a, l2b, l3a, l3b, 64);
    ssim(l3a, l3b, 64, 3);
    down(l3a, l3b, l4a, l4b, 32);
    ssim(l4a, l4b, 32, 4);

    finalize_kernel<<<1, 1, 0, stream>>>(acc, out);
}